// HebbianConv2d_15161234555377
// MI455X (gfx1250) — compile-verified
//
#include <hip/hip_runtime.h>
#include <hip/hip_bf16.h>

// ---------------------------------------------------------------------------
// HebbianConv2d for MI455X (gfx1250, wave32, WMMA).
//  x      : [32, 64, 64, 64]  fp32 (NCHW)
//  weight : [128, 64, 3, 3]   fp32
//  bias   : [128]             fp32
//  out    : y [32,128,62,62] fp32  ++  weight_update [128,64,3,3] fp32
// Conv as implicit GEMM with V_WMMA_F32_16X16X4_F32 (fp32 matrix pipe keeps
// argmax winners bit-faithful). Hebbian WTA reduction uses a full 294 KB
// per-workgroup LDS accumulator (only possible with CDNA5's 320 KB LDS).
// ---------------------------------------------------------------------------

typedef float v2f __attribute__((ext_vector_type(2)));
typedef float v8f __attribute__((ext_vector_type(8)));

#define BATCH   32
#define CIN     64
#define COUT    128
#define IMH     64
#define IMW     64
#define KH      3
#define KW      3
#define OH      62
#define OW      62
#define LOC     (OH * OW)            // 3844 output locations per image
#define M_TOT   (BATCH * LOC)        // 123008 GEMM rows
#define KRED    (CIN * KH * KW)      // 576 reduction length
#define WU_ELEMS (COUT * KRED)       // 73728 weight-update elements
#define Y_ELEMS (BATCH * COUT * LOC)

#define LDS_STRIDE 578               // 576 padded: bank-spread + 8B-aligned rows

// ---------------------------------------------------------------------------
// Kernel 1: weight normalization  wn[k][:] = w[k][:] / ||w[k]|| (guard 0)
// ---------------------------------------------------------------------------
__global__ __launch_bounds__(256)
void hc_normalize_w(const float* __restrict__ w, float* __restrict__ wn) {
    __shared__ float red[256];
    const int k = blockIdx.x;
    const float* wk = w + k * KRED;
    float s = 0.f;
    for (int i = threadIdx.x; i < KRED; i += 256) {
        float v = wk[i];
        s += v * v;
    }
    red[threadIdx.x] = s;
    __syncthreads();
    for (int off = 128; off > 0; off >>= 1) {
        if (threadIdx.x < (unsigned)off) red[threadIdx.x] += red[threadIdx.x + off];
        __syncthreads();
    }
    const float nrm = sqrtf(red[0]);
    const float inv = (nrm == 0.f) ? 1.f : (1.f / nrm);
    for (int i = threadIdx.x; i < KRED; i += 256) {
        wn[k * KRED + i] = wk[i] * inv;
    }
}

// ---------------------------------------------------------------------------
// Kernel 2: zero fp32 buffer
// ---------------------------------------------------------------------------
__global__ __launch_bounds__(256)
void hc_zero_f32(float* __restrict__ p, int n) {
    int i = blockIdx.x * 256 + threadIdx.x;
    if (i < n) p[i] = 0.f;
}

// ---------------------------------------------------------------------------
// Kernel 3: conv via implicit GEMM, V_WMMA_F32_16X16X4_F32.
// Block = 256 thr = 8 waves. Block tile: M=32, N=128 (wave owns N=16, two
// M=16 subtiles sharing each B fragment). A tile (32x576) staged in LDS.
// WMMA f32 16x16x4 fragment layout (wave32):
//   A (16x4): lane l<16  holds A[M=l][K=0],A[l][1];  lane l>=16 A[l-16][2],[3]
//   B (4x16): lane l<16  holds B[K=0][N=l],B[1][l];  lane l>=16 B[2][l-16],[3][..]
//   C/D     : vgpr r, lane l<16 -> (M=r, N=l); lane l>=16 -> (M=r+8, N=l-16)
// ---------------------------------------------------------------------------
__global__ __launch_bounds__(256)
void hc_conv_wmma(const float* __restrict__ x, const float* __restrict__ wn,
                  const float* __restrict__ bias, float* __restrict__ y) {
    extern __shared__ float atile[];               // 32 * LDS_STRIDE floats
    const int tid = threadIdx.x;
    const int m0  = blockIdx.x * 32;

    // ---- cooperative im2col load: 32 rows x 576 cols of patches into LDS ----
    for (int idx = tid; idx < 32 * KRED; idx += 256) {
        const int r   = idx / KRED;
        const int kp  = idx - r * KRED;
        const int m   = m0 + r;
        const int b   = m / LOC;
        const int rem = m - b * LOC;
        const int oh  = rem / OW;
        const int ow  = rem - oh * OW;
        const int c   = kp / (KH * KW);
        const int r2  = kp - c * (KH * KW);
        const int ki  = r2 / KW;
        const int kj  = r2 - ki * KW;
        atile[r * LDS_STRIDE + kp] =
            x[((b * CIN + c) * IMH + (oh + ki)) * IMW + (ow + kj)];
    }
    __syncthreads();

    const int lane  = tid & 31;
    const int wave  = tid >> 5;        // 0..7 -> N strip
    const int nbase = wave * 16;
    const int lhalf = lane >> 4;       // 0 | 1 : K pair selector
    const int lmod  = lane & 15;

    v8f acc0 = {};                     // rows m0 .. m0+15
    v8f acc1 = {};                     // rows m0+16 .. m0+31

    const float* a0p  = &atile[lmod * LDS_STRIDE + 2 * lhalf];
    const float* a1p  = &atile[(16 + lmod) * LDS_STRIDE + 2 * lhalf];
    const float* wptr = wn + (nbase + lmod) * KRED + 2 * lhalf;

    for (int kk = 0; kk < KRED; kk += 4) {
        v2f a0 = *(const v2f*)(a0p + kk);          // ds_load_b64
        v2f a1 = *(const v2f*)(a1p + kk);          // ds_load_b64
        v2f bb = *(const v2f*)(wptr + kk);         // global_load_b64 (L2 hot)
        acc0 = __builtin_amdgcn_wmma_f32_16x16x4_f32(
                   false, a0, false, bb, (short)0, acc0, false, false);
        acc1 = __builtin_amdgcn_wmma_f32_16x16x4_f32(
                   false, a1, false, bb, (short)0, acc1, false, false);
    }

    // ---- epilogue: bias add + scatter D tiles to y[NCHW] ----
    const int n  = nbase + lmod;
    const float bv = bias[n];
#pragma unroll
    for (int r = 0; r < 8; ++r) {
        {   // acc0: row = r + 8*lhalf
            const int m   = m0 + r + 8 * lhalf;
            const int b   = m / LOC;
            const int rem = m - b * LOC;
            y[(b * COUT + n) * LOC + rem] = acc0[r] + bv;
        }
        {   // acc1: row = 16 + r + 8*lhalf
            const int m   = m0 + 16 + r + 8 * lhalf;
            const int b   = m / LOC;
            const int rem = m - b * LOC;
            y[(b * COUT + n) * LOC + rem] = acc1[r] + bv;
        }
    }
}

// ---------------------------------------------------------------------------
// Kernel 4: WTA Hebbian accumulation.
// Each block holds the FULL 128x576 fp32 accumulator (294,912 B) in LDS —
// fits CDNA5's 320 KB WGP LDS. Per location: argmax over 128 channels of y,
// then 576 ds_add_f32 into the winner's row. One agent-scope global fadd
// flush per block at the end.
// ---------------------------------------------------------------------------
#define HB_BLOCKS 128

__global__ __launch_bounds__(256)
void hc_hebb_accum(const float* __restrict__ x, const float* __restrict__ y,
                   float* __restrict__ hebb) {
    extern __shared__ float hl[];                  // WU_ELEMS floats
    const int tid = threadIdx.x;

    for (int i = tid; i < WU_ELEMS; i += 256) hl[i] = 0.f;
    __syncthreads();

    for (int m = blockIdx.x * 256 + tid; m < M_TOT; m += HB_BLOCKS * 256) {
        const int b   = m / LOC;
        const int rem = m - b * LOC;
        const int oh  = rem / OW;
        const int ow  = rem - oh * OW;

        // argmax over channels (first max wins, matching jnp.argmax)
        const float* yp = y + (size_t)b * COUT * LOC + rem;
        int   best = 0;
        float bv   = yp[0];
        for (int k = 1; k < COUT; ++k) {
            const float v = yp[(size_t)k * LOC];
            if (v > bv) { bv = v; best = k; }
        }

        // accumulate this location's patch into the winner's LDS row
        const float* xp = x + (size_t)b * CIN * IMH * IMW;
        float* hp = hl + best * KRED;
        for (int c = 0; c < CIN; ++c) {
            const float* xc = xp + c * IMH * IMW + oh * IMW + ow;
#pragma unroll
            for (int ki = 0; ki < KH; ++ki) {
#pragma unroll
                for (int kj = 0; kj < KW; ++kj) {
                    __hip_atomic_fetch_add(&hp[c * 9 + ki * 3 + kj],
                                           xc[ki * IMW + kj],
                                           __ATOMIC_RELAXED,
                                           __HIP_MEMORY_SCOPE_WORKGROUP);
                }
            }
        }
    }
    __syncthreads();

    // flush block-private accumulator to global
    for (int i = tid; i < WU_ELEMS; i += 256) {
        const float v = hl[i];
        if (v != 0.f) {
            __hip_atomic_fetch_add(&hebb[i], v, __ATOMIC_RELAXED,
                                   __HIP_MEMORY_SCOPE_AGENT);
        }
    }
}

// ---------------------------------------------------------------------------
// Kernel 5: weight_update = hebb / (B*L + 1e-8)
// ---------------------------------------------------------------------------
__global__ __launch_bounds__(256)
void hc_finalize(const float* __restrict__ hebb, float* __restrict__ wu, int n) {
    const int i = blockIdx.x * 256 + threadIdx.x;
    if (i < n) {
        const float scale = 1.0f / ((float)M_TOT + 1e-8f);
        wu[i] = hebb[i] * scale;
    }
}

// ---------------------------------------------------------------------------
extern "C" void kernel_launch(void* const* d_in, const int* in_sizes, int n_in,
                              void* d_out, int out_size, void* d_ws, size_t ws_size,
                              hipStream_t stream) {
    const float* x      = (const float*)d_in[0];
    const float* weight = (const float*)d_in[1];
    const float* bias   = (const float*)d_in[2];

    float* y  = (float*)d_out;                    // [32,128,62,62]
    float* wu = (float*)d_out + Y_ELEMS;          // [128,64,3,3]

    float* wn   = (float*)d_ws;                   // normalized weights [128][576]
    float* hebb = wn + WU_ELEMS;                  // global hebb accumulator

    // 1) normalize weights (one block per out-channel)
    hc_normalize_w<<<COUT, 256, 0, stream>>>(weight, wn);

    // 2) zero hebb accumulator (fresh every call — deterministic)
    hc_zero_f32<<<(WU_ELEMS + 255) / 256, 256, 0, stream>>>(hebb, WU_ELEMS);

    // 3) conv via fp32 WMMA implicit GEMM: 123008/32 = 3844 blocks
    hc_conv_wmma<<<M_TOT / 32, 256, 32 * LDS_STRIDE * sizeof(float), stream>>>(
        x, wn, bias, y);

    // 4) WTA Hebbian reduction: full accumulator in 294,912 B of LDS
    hc_hebb_accum<<<HB_BLOCKS, 256, WU_ELEMS * sizeof(float), stream>>>(
        x, y, hebb);

    // 5) scale into output
    hc_finalize<<<(WU_ELEMS + 255) / 256, 256, 0, stream>>>(hebb, wu, WU_ELEMS);
}